// GNN_infer_51539607552523
// MI455X (gfx1250) — compile-verified
//
#include <hip/hip_runtime.h>
#include <math.h>

// ---------------------------------------------------------------------------
// GNN_infer for MI455X (gfx1250, wave32) -- single fused kernel.
// B=4, H=W=128 -> 65536 pixels; HID=10; IN=256.
//
// Roofline: ~190MB unavoidable HBM traffic @ 23.3TB/s ~ 8.2us; compute ~1.7GF
// is negligible => eliminate ALL avoidable traffic. One workgroup (256 thr,
// 8 waves) owns 256 consecutive pixels:
//   phase 0: stage padded/zero-extended weight panel [2][256][16] into LDS,
//            pre-swizzled into per-lane WMMA-B float2 fragments
//   phase 1: 32 wave-tasks (16 pixel-tiles x {h_fea,p_fea}), each a 16x16
//            output tile via 64x v_wmma_f32_16x16x4_f32 over K=256;
//            chunked 8x8 so every A load is base+24-bit-immediate;
//            results -> LDS [fea][pix][16]  (no global intermediate)
//   phase 2: per-thread fused graph message passing, all outputs written once
// ---------------------------------------------------------------------------

typedef float v2f __attribute__((ext_vector_type(2)));
typedef float v8f __attribute__((ext_vector_type(8)));

#define CHS   16384            // channel stride (H*W)
#define XBS   163840           // batch stride of [B,10,H,W] tensors
#define FBS   4194304          // batch stride of [B,256,H,W] tensors

// d_out offsets in floats (reference tuple order, flat-concatenated)
#define OUT_XP     0
#define OUT_XH     3932160
#define OUT_XF     5242880
#define OUT_HPATT  5898240
#define OUT_FHATT  6291456
#define OUT_XFMSG  6422528

__device__ __forceinline__ float sigmoidf(float x) {
    return 1.0f / (1.0f + __expf(-x));
}

struct Params {
    const float *xp, *xh, *xf, *h_fea, *p_fea;
    const float *fg_comp_W, *fg_comp_g, *fg_comp_b;
    const float *fg_up_W, *fg_up_g, *fg_up_b;
    const float *hg_dec_Watt, *hg_dec_batt, *hg_dec_Wc, *hg_dec_g, *hg_dec_b;
    const float *hg_compu_W, *hg_compu_g, *hg_compu_b;
    const float *hg_compl_W, *hg_compl_g, *hg_compl_b;
    const float *hg_pdp_W, *hg_pdp_g, *hg_pdp_b;
    const float *hg_att_W, *hg_att_b;
    const float *hg_up_W, *hg_up_g, *hg_up_b;
    const float *pg_pdp_W, *pg_pdp_g, *pg_pdp_b;
    const float *pg_att_W, *pg_att_b;
    const float *pg_dec_Watt, *pg_dec_batt, *pg_dec_Wc, *pg_dec_g, *pg_dec_b;
    const float *pg_up_W, *pg_up_g, *pg_up_b;
    float *out;
};

// Padded weight panel element WcombT[fea][c][n]:
//  fea=0 (h_fea): n<10 -> hg_pdp_W[n][c](row len 266); n=10,11 -> hg_dec_Watt[n-10][c](276)
//  fea=1 (p_fea): n<10 -> pg_pdp_W[n][c];              n=10..15 -> pg_dec_Watt[n-10][c]
__device__ __forceinline__ float panel_val(const Params& P, int fea, int c, int n) {
    if (fea == 0) {
        if (n < 10) return P.hg_pdp_W[n * 266 + c];
        if (n < 12) return P.hg_dec_Watt[(n - 10) * 276 + c];
        return 0.0f;
    } else {
        if (n < 10) return P.pg_pdp_W[n * 266 + c];
        return P.pg_dec_Watt[(n - 10) * 276 + c];
    }
}

__global__ void __launch_bounds__(256) gnn_fused(Params P) {
    // LDS: results [2][256][16] floats (32KB) + B-fragment pairs
    //      [2][64 kk][32 lane] float2 (32KB)
    __shared__ float ldsR[2 * 256 * 16];
    __shared__ v2f   ldsW[2 * 64 * 32];

    const int tid  = threadIdx.x;
    const int lane = tid & 31;
    const int wave = tid >> 5;
    const int m     = lane & 15;          // pixel row within tile / weight col
    const int koff  = (lane >> 4) << 1;   // 0 or 2
    const int mBase = (lane >> 4) << 3;   // 0 or 8

    const int pixBlock = blockIdx.x * 256;         // 256 pixels per block
    const int b      = pixBlock >> 14;             // whole block in one batch
    const int hwBase = pixBlock & 16383;

    // ---------------- phase 0: stage swizzled B fragments ----------------
#pragma unroll
    for (int q = 0; q < 16; ++q) {
        const int e   = tid * 16 + q;              // 0..4095
        const int fea = e >> 11;
        const int rem = e & 2047;
        const int kk  = rem >> 5;
        const int ln  = rem & 31;
        const int c0  = kk * 4 + ((ln >> 4) << 1);
        const int n   = ln & 15;
        v2f w;
        w.x = panel_val(P, fea, c0, n);
        w.y = panel_val(P, fea, c0 + 1, n);
        ldsW[e] = w;
    }
    __syncthreads();

    // ---------------- phase 1: WMMA GEMMs -> LDS ----------------
#pragma unroll
    for (int q = 0; q < 4; ++q) {
        const int task = wave * 4 + q;             // 0..31
        const int fea  = task >> 4;
        const int tile = task & 15;
        const float* __restrict__ src = fea ? P.p_fea : P.h_fea;

        // per-lane A base: pixel (hwBase+tile*16+m) of batch b, channel koff
        const float* __restrict__ aptr =
            src + b * FBS + hwBase + tile * 16 + m + koff * CHS;
        const v2f* wp = ldsW + fea * 2048 + lane;  // + kk*32

        v8f acc = {};
        // 8 chunks x 8 k-steps: inside a chunk the largest load offset is
        // 29*CHS*4 = 1.9MB < 8MB, so every A load is base + 24-bit immediate.
        for (int chunk = 0; chunk < 8; ++chunk) {
#pragma unroll
            for (int u = 0; u < 8; ++u) {
                v2f a;
                a.x = aptr[(u * 4) * CHS];
                a.y = aptr[(u * 4 + 1) * CHS];
                v2f bb = wp[u * 32];
                acc = __builtin_amdgcn_wmma_f32_16x16x4_f32(
                          false, a, false, bb, (short)0, acc, false, false);
            }
            aptr += 32 * CHS;   // one 64-bit pointer bump per 8 WMMAs
            wp   += 8 * 32;
        }
        float* r = ldsR + fea * 4096 + (tile * 16 + mBase) * 16 + m;
#pragma unroll
        for (int rI = 0; rI < 8; ++rI) r[rI * 16] = acc[rI];
    }
    __syncthreads();

    // ---------------- phase 2: per-pixel fused graph ----------------
    const int p  = pixBlock + tid;
    const int hw = p & 16383;
    const int xoff = b * XBS + hw;
    float* __restrict__ out = P.out;

    float xf[10], xh0[10], xh1[10];
#pragma unroll
    for (int c = 0; c < 10; ++c) {
        xf[c]  = P.xf[xoff + c * CHS];
        xh0[c] = P.xh[xoff + c * CHS];
        xh1[c] = P.xh[655360 + xoff + c * CHS];
    }
    float hgBase[10], pgBase[10], pAttF[6];
#pragma unroll
    for (int c = 0; c < 10; ++c) {
        hgBase[c] = ldsR[tid * 16 + c];
        pgBase[c] = ldsR[4096 + tid * 16 + c];
    }
    const float hAttF0 = ldsR[tid * 16 + 10];
    const float hAttF1 = ldsR[tid * 16 + 11];
#pragma unroll
    for (int i = 0; i < 6; ++i) pAttF[i] = ldsR[4096 + tid * 16 + 10 + i];

    // ----------------- full graph -----------------
    float msg[10];
#pragma unroll
    for (int c = 0; c < 10; ++c) {
        float acc = 0.f;
#pragma unroll
        for (int j = 0; j < 10; ++j) {
            acc = fmaf(P.fg_comp_W[c * 30 + j],      xf[j],  acc);
            acc = fmaf(P.fg_comp_W[c * 30 + 10 + j], xh0[j], acc);
            acc = fmaf(P.fg_comp_W[c * 30 + 20 + j], xh1[j], acc);
        }
        float v = fmaxf(fmaf(acc, P.fg_comp_g[c], P.fg_comp_b[c]), 0.f);
        msg[c] = v;
        out[OUT_XFMSG + b * XBS + c * CHS + hw] = v;
    }
#pragma unroll
    for (int c = 0; c < 10; ++c) {
        float acc = 0.f;
#pragma unroll
        for (int j = 0; j < 10; ++j) {
            acc = fmaf(P.fg_up_W[c * 20 + j],      xf[j],  acc);
            acc = fmaf(P.fg_up_W[c * 20 + 10 + j], msg[j], acc);
        }
        out[OUT_XF + b * XBS + c * CHS + hw] =
            fmaxf(xf[c] + fmaf(acc, P.fg_up_g[c], P.fg_up_b[c]), 0.f);
    }

    // ----------------- half graph -----------------
    float hdp0[10], hdp1[10];
#pragma unroll
    for (int c = 0; c < 10; ++c) {
        float a0 = hgBase[c], a1 = hgBase[c];
#pragma unroll
        for (int j = 0; j < 10; ++j) {
            const float w = P.hg_pdp_W[c * 266 + 256 + j];
            a0 = fmaf(w, xh0[j], a0);
            a1 = fmaf(w, xh1[j], a1);
        }
        hdp0[c] = fmaxf(fmaf(a0, P.hg_pdp_g[c], P.hg_pdp_b[c]), 0.f);
        hdp1[c] = fmaxf(fmaf(a1, P.hg_pdp_g[c], P.hg_pdp_b[c]), 0.f);
    }
    float ha0 = P.hg_att_b[0], ha1 = P.hg_att_b[1];
#pragma unroll
    for (int j = 0; j < 10; ++j) {
        ha0 = fmaf(P.hg_att_W[j],      xh0[j], ha0);
        ha1 = fmaf(P.hg_att_W[10 + j], xh1[j], ha1);
    }
    ha0 = sigmoidf(ha0); ha1 = sigmoidf(ha1);

    float su = hAttF0 + P.hg_dec_batt[0];
    float sl = hAttF1 + P.hg_dec_batt[1];
#pragma unroll
    for (int j = 0; j < 10; ++j) {
        su = fmaf(P.hg_dec_Watt[256 + j],       xf[j],  su);
        su = fmaf(P.hg_dec_Watt[266 + j],       xh0[j], su);
        sl = fmaf(P.hg_dec_Watt[276 + 256 + j], xf[j],  sl);
        sl = fmaf(P.hg_dec_Watt[276 + 266 + j], xh1[j], sl);
    }
    const float att_u = sigmoidf(su), att_l = sigmoidf(sl);
    out[OUT_FHATT + b * 32768 + hw]       = att_u;
    out[OUT_FHATT + b * 32768 + CHS + hw] = att_l;

    float xp[6][10];
#pragma unroll
    for (int i = 0; i < 6; ++i)
#pragma unroll
        for (int c = 0; c < 10; ++c)
            xp[i][c] = P.xp[i * 655360 + xoff + c * CHS];

    // upper half update
    float mU[10];
#pragma unroll
    for (int c = 0; c < 10; ++c) {
        float acc = 0.f;
#pragma unroll
        for (int j = 0; j < 10; ++j) acc = fmaf(P.hg_compu_W[c * 50 + j], xh0[j], acc);
#pragma unroll
        for (int i = 0; i < 4; ++i)
#pragma unroll
            for (int j = 0; j < 10; ++j)
                acc = fmaf(P.hg_compu_W[c * 50 + 10 + i * 10 + j], xp[i][j], acc);
        const float xphu = fmaxf(fmaf(acc, P.hg_compu_g[c], P.hg_compu_b[c]), 0.f);
        const float xlh  = hdp1[c] * ha1 * (1.f - ha0);
        float a2 = 0.f;
#pragma unroll
        for (int j = 0; j < 10; ++j) a2 = fmaf(P.hg_dec_Wc[c * 10 + j], xf[j], a2);
        const float xfhu = fmaxf(fmaf(a2, P.hg_dec_g[c], P.hg_dec_b[c]), 0.f) * att_u;
        mU[c] = fmaxf(fmaxf(xphu, xlh), xfhu);
    }
#pragma unroll
    for (int c = 0; c < 10; ++c) {
        float acc = 0.f;
#pragma unroll
        for (int j = 0; j < 10; ++j) {
            acc = fmaf(P.hg_up_W[c * 20 + j],      xh0[j], acc);
            acc = fmaf(P.hg_up_W[c * 20 + 10 + j], mU[j],  acc);
        }
        out[OUT_XH + b * XBS + c * CHS + hw] =
            fmaxf(xh0[c] + fmaf(acc, P.hg_up_g[c], P.hg_up_b[c]), 0.f);
    }
    // lower half update
    float mL[10];
#pragma unroll
    for (int c = 0; c < 10; ++c) {
        float acc = 0.f;
#pragma unroll
        for (int j = 0; j < 10; ++j) acc = fmaf(P.hg_compl_W[c * 30 + j], xh1[j], acc);
#pragma unroll
        for (int i = 0; i < 2; ++i)
#pragma unroll
            for (int j = 0; j < 10; ++j)
                acc = fmaf(P.hg_compl_W[c * 30 + 10 + i * 10 + j], xp[4 + i][j], acc);
        const float xphl = fmaxf(fmaf(acc, P.hg_compl_g[c], P.hg_compl_b[c]), 0.f);
        const float xuh  = hdp0[c] * ha0 * (1.f - ha1);
        float a2 = 0.f;
#pragma unroll
        for (int j = 0; j < 10; ++j) a2 = fmaf(P.hg_dec_Wc[100 + c * 10 + j], xf[j], a2);
        const float xfhl = fmaxf(fmaf(a2, P.hg_dec_g[10 + c], P.hg_dec_b[10 + c]), 0.f) * att_l;
        mL[c] = fmaxf(fmaxf(xphl, xuh), xfhl);
    }
#pragma unroll
    for (int c = 0; c < 10; ++c) {
        float acc = 0.f;
#pragma unroll
        for (int j = 0; j < 10; ++j) {
            acc = fmaf(P.hg_up_W[200 + c * 20 + j],      xh1[j], acc);
            acc = fmaf(P.hg_up_W[200 + c * 20 + 10 + j], mL[j],  acc);
        }
        out[OUT_XH + 655360 + b * XBS + c * CHS + hw] =
            fmaxf(xh1[c] + fmaf(acc, P.hg_up_g[10 + c], P.hg_up_b[10 + c]), 0.f);
    }

    // ----------------- part graph -----------------
    float satt[6], pdp[6][10];
#pragma unroll
    for (int i = 0; i < 6; ++i) {
        float s = P.pg_att_b[i];
#pragma unroll
        for (int j = 0; j < 10; ++j) s = fmaf(P.pg_att_W[i * 10 + j], xp[i][j], s);
        satt[i] = sigmoidf(s);
#pragma unroll
        for (int c = 0; c < 10; ++c) {
            float a = pgBase[c];
#pragma unroll
            for (int j = 0; j < 10; ++j)
                a = fmaf(P.pg_pdp_W[c * 266 + 256 + j], xp[i][j], a);
            pdp[i][c] = fmaxf(fmaf(a, P.pg_pdp_g[c], P.pg_pdp_b[c]), 0.f);
        }
    }
#pragma unroll
    for (int t = 0; t < 6; ++t) {
        const int s0 = (t == 0) ? 1 : t - 1;
        const int s1 = (t == 0 || t == 5) ? -1 : t + 1;
        float sH[10];
#pragma unroll
        for (int j = 0; j < 10; ++j) sH[j] = (t < 4) ? xh0[j] : xh1[j];
        float s = pAttF[t] + P.pg_dec_batt[t];
#pragma unroll
        for (int j = 0; j < 10; ++j) {
            s = fmaf(P.pg_dec_Watt[t * 276 + 256 + j], sH[j],    s);
            s = fmaf(P.pg_dec_Watt[t * 276 + 266 + j], xp[t][j], s);
        }
        const float patt = sigmoidf(s);
        out[OUT_HPATT + b * 98304 + t * CHS + hw] = 0.5f * (satt[t] + patt);
        float mm[10];
#pragma unroll
        for (int c = 0; c < 10; ++c) {
            float m2 = pdp[s0][c] * satt[s0] * (1.f - satt[t]);
            if (s1 >= 0) m2 = fmaxf(m2, pdp[s1][c] * satt[s1] * (1.f - satt[t]));
            float a = 0.f;
#pragma unroll
            for (int j = 0; j < 10; ++j)
                a = fmaf(P.pg_dec_Wc[t * 100 + c * 10 + j], sH[j], a);
            const float xhp =
                fmaxf(fmaf(a, P.pg_dec_g[t * 10 + c], P.pg_dec_b[t * 10 + c]), 0.f) * patt;
            mm[c] = fmaxf(m2, xhp);
        }
#pragma unroll
        for (int c = 0; c < 10; ++c) {
            float acc = 0.f;
#pragma unroll
            for (int j = 0; j < 10; ++j) {
                acc = fmaf(P.pg_up_W[t * 200 + c * 20 + j],      xp[t][j], acc);
                acc = fmaf(P.pg_up_W[t * 200 + c * 20 + 10 + j], mm[j],    acc);
            }
            out[OUT_XP + t * 655360 + b * XBS + c * CHS + hw] =
                fmaxf(xp[t][c] + fmaf(acc, P.pg_up_g[t * 10 + c], P.pg_up_b[t * 10 + c]), 0.f);
        }
    }
}

// ---------------------------------------------------------------------------
extern "C" void kernel_launch(void* const* d_in, const int* in_sizes, int n_in,
                              void* d_out, int out_size, void* d_ws, size_t ws_size,
                              hipStream_t stream) {
    // setup_inputs() order: xp, xh, xf, bg_node, p_fea, h_fea, f_fea, then
    // params flattened in dict order.
    Params P;
    P.xp    = (const float*)d_in[0];
    P.xh    = (const float*)d_in[1];
    P.xf    = (const float*)d_in[2];
    P.p_fea = (const float*)d_in[4];
    P.h_fea = (const float*)d_in[5];
    int i = 7;
    P.fg_comp_W  = (const float*)d_in[i++];
    P.fg_comp_g  = (const float*)d_in[i++];
    P.fg_comp_b  = (const float*)d_in[i++];
    P.fg_up_W    = (const float*)d_in[i++];
    P.fg_up_g    = (const float*)d_in[i++];
    P.fg_up_b    = (const float*)d_in[i++];
    P.hg_dec_Watt = (const float*)d_in[i++];
    P.hg_dec_batt = (const float*)d_in[i++];
    P.hg_dec_Wc   = (const float*)d_in[i++];
    P.hg_dec_g    = (const float*)d_in[i++];
    P.hg_dec_b    = (const float*)d_in[i++];
    P.hg_compu_W  = (const float*)d_in[i++];
    P.hg_compu_g  = (const float*)d_in[i++];
    P.hg_compu_b  = (const float*)d_in[i++];
    P.hg_compl_W  = (const float*)d_in[i++];
    P.hg_compl_g  = (const float*)d_in[i++];
    P.hg_compl_b  = (const float*)d_in[i++];
    P.hg_pdp_W    = (const float*)d_in[i++];
    P.hg_pdp_g    = (const float*)d_in[i++];
    P.hg_pdp_b    = (const float*)d_in[i++];
    P.hg_att_W    = (const float*)d_in[i++];
    P.hg_att_b    = (const float*)d_in[i++];
    P.hg_up_W     = (const float*)d_in[i++];
    P.hg_up_g     = (const float*)d_in[i++];
    P.hg_up_b     = (const float*)d_in[i++];
    P.pg_pdp_W    = (const float*)d_in[i++];
    P.pg_pdp_g    = (const float*)d_in[i++];
    P.pg_pdp_b    = (const float*)d_in[i++];
    P.pg_att_W    = (const float*)d_in[i++];
    P.pg_att_b    = (const float*)d_in[i++];
    P.pg_dec_Watt = (const float*)d_in[i++];
    P.pg_dec_batt = (const float*)d_in[i++];
    P.pg_dec_Wc   = (const float*)d_in[i++];
    P.pg_dec_g    = (const float*)d_in[i++];
    P.pg_dec_b    = (const float*)d_in[i++];
    P.pg_up_W     = (const float*)d_in[i++];
    P.pg_up_g     = (const float*)d_in[i++];
    P.pg_up_b     = (const float*)d_in[i++];
    P.out = (float*)d_out;

    gnn_fused<<<256, 256, 0, stream>>>(P);
}